// SGAT_57303453663284
// MI455X (gfx1250) — compile-verified
//
#include <hip/hip_runtime.h>
#include <hip/hip_bf16.h>
#include <math.h>

#define N_NODES 100000
#define N_EDGES 1600000
#define IN_DIM  128
#define HID_DIM 64
#define OUT_DIM 128
#define QKV_DIM 256   // fused columns: [q(64) | k(64) | v(128)]
#define EPS     1e-5f

typedef __attribute__((ext_vector_type(2))) float v2f;
typedef __attribute__((ext_vector_type(8))) float v8f;

// ---- order-preserving float <-> uint map for atomicMax-based segment_max ----
__device__ __forceinline__ unsigned enc_f32(float f) {
    unsigned u = __float_as_uint(f);
    return (u & 0x80000000u) ? ~u : (u | 0x80000000u);
}
__device__ __forceinline__ float dec_f32(unsigned u) {
    return (u & 0x80000000u) ? __uint_as_float(u & 0x7FFFFFFFu)
                             : __uint_as_float(~u);
}
__device__ __forceinline__ float sigmoidf_(float x) {
    return 1.0f / (1.0f + __expf(-x));
}

// ---------------- kernel 1: zero output + stats/segment scratch -------------
__global__ void k_init(float* __restrict__ out, float* __restrict__ wspre) {
    size_t i      = (size_t)blockIdx.x * blockDim.x + threadIdx.x;
    size_t stride = (size_t)gridDim.x * blockDim.x;
    for (size_t j = i; j < (size_t)N_NODES * OUT_DIM; j += stride) out[j] = 0.f;
    for (size_t j = i; j < (size_t)(512 + 2 * N_NODES); j += stride) wspre[j] = 0.f;
}

// ---------------- kernel 2: per-column sum / sum-of-squares -----------------
__global__ void k_colstats(const float* __restrict__ feat,
                           float* __restrict__ colsum,
                           float* __restrict__ colsumsq) {
    int c = threadIdx.x;                       // 0..127, one column per thread
    float s = 0.f, s2 = 0.f;
    for (int row = blockIdx.x; row < N_NODES; row += gridDim.x) {
        float v = feat[(size_t)row * IN_DIM + c];
        s += v; s2 += v * v;
    }
    atomicAdd(&colsum[c], s);
    atomicAdd(&colsumsq[c], s2);
}

// ---------------- kernel 3: fold mean/var/gamma/beta into scale/shift -------
__global__ void k_stats_fin(const float* __restrict__ colsum,
                            const float* __restrict__ colsumsq,
                            const float* __restrict__ gamma,
                            const float* __restrict__ beta,
                            float* __restrict__ scale,
                            float* __restrict__ shift) {
    int c = threadIdx.x;
    float mean = colsum[c]   * (1.0f / (float)N_NODES);
    float var  = colsumsq[c] * (1.0f / (float)N_NODES) - mean * mean;
    float rs   = rsqrtf(var + EPS) * gamma[c];
    scale[c] = rs;
    shift[c] = beta[c] - mean * rs;
}

// ---------------- kernel 4: fused norm + q|k|v GEMM via V_WMMA_F32_16X16X4 --
// 512 threads = 16 waves. Block computes a 16x256 tile of Y = norm(X) @ [Wq|Wk|Wv].
// Each wave owns one 16-col tile; K=128 consumed in 32 steps of 4.
// Fragment layouts per CDNA5 ISA 7.12.2 (32-bit A 16x4, 32-bit C/D 16x16).
__global__ void __launch_bounds__(512)
k_qkv_wmma(const float* __restrict__ feat, const float* __restrict__ scale,
           const float* __restrict__ shift, const float* __restrict__ Wq,
           const float* __restrict__ bq, const float* __restrict__ Wk,
           const float* __restrict__ Wv, float* __restrict__ Y) {
    __shared__ float xs[16][132];              // pad 128->132: conflict-free col reads
    const int    tid     = threadIdx.x;
    const size_t rowbase = (size_t)blockIdx.x * 16;   // 100000 = 6250*16 exactly

    for (int i = tid; i < 16 * 128; i += 512) {       // stage + normalize A tile
        int r = i >> 7, c = i & 127;
        xs[r][c] = feat[(rowbase + r) * IN_DIM + c] * scale[c] + shift[c];
    }
    __syncthreads();

    const int wave    = tid >> 5;
    const int lane    = tid & 31;
    const int half    = lane >> 4;             // 0: K pair {k,k+1}, 1: {k+2,k+3}
    const int mrow    = lane & 15;             // A row / B,D column within tile
    const int colbase = wave * 16;

    const float* wsrc; int ldw, wc;
    if (wave < 4)      { wsrc = Wq; ldw = 64;  wc = colbase        + mrow; }
    else if (wave < 8) { wsrc = Wk; ldw = 64;  wc = colbase - 64   + mrow; }
    else               { wsrc = Wv; ldw = 128; wc = colbase - 128  + mrow; }

    v8f acc = {};
#pragma unroll
    for (int kk = 0; kk < IN_DIM; kk += 4) {
        int k0 = kk + half * 2;
        v2f a, b;
        a.x = xs[mrow][k0];
        a.y = xs[mrow][k0 + 1];
        b.x = wsrc[(size_t)k0 * ldw + wc];
        b.y = wsrc[(size_t)(k0 + 1) * ldw + wc];
        acc = __builtin_amdgcn_wmma_f32_16x16x4_f32(
            /*neg_a=*/false, a, /*neg_b=*/false, b,
            /*c_mod=*/(short)0, acc, /*reuse_a=*/false, /*reuse_b=*/false);
    }

    float bias = (wave < 4) ? bq[wc] : 0.f;
#pragma unroll
    for (int r = 0; r < 8; ++r) {              // D: VGPR r -> row r + half*8
        int M = r + half * 8;
        Y[(rowbase + M) * QKV_DIM + colbase + mrow] = acc[r] + bias;
    }
}

// ---------------- kernel 5: e = sigmoid(q[src]+k[dst]) @ We ; segment max ---
__global__ void k_edge_logits(const int* __restrict__ src, const int* __restrict__ dst,
                              const float* __restrict__ Y, const float* __restrict__ We,
                              float* __restrict__ e_out, unsigned* __restrict__ m_enc) {
    int i = blockIdx.x * blockDim.x + threadIdx.x;
    if (i >= N_EDGES) return;
    int s = src[i], d = dst[i];
    const float4* qp = (const float4*)(Y + (size_t)s * QKV_DIM);        // q @ 0
    const float4* kp = (const float4*)(Y + (size_t)d * QKV_DIM + 64);   // k @ 64
    const float4* wp = (const float4*)We;
    float acc = 0.f;
#pragma unroll
    for (int h = 0; h < HID_DIM / 4; ++h) {
        float4 q4 = qp[h], k4 = kp[h], w4 = wp[h];
        acc += sigmoidf_(q4.x + k4.x) * w4.x;
        acc += sigmoidf_(q4.y + k4.y) * w4.y;
        acc += sigmoidf_(q4.z + k4.z) * w4.z;
        acc += sigmoidf_(q4.w + k4.w) * w4.w;
    }
    e_out[i] = acc;
    atomicMax(&m_enc[d], enc_f32(acc));
}

// ---------------- kernel 6: wave-per-edge softmax-weighted scatter ----------
// out[dst] += exp(e-m[dst]) * v[src] ; s[dst] += exp(e-m[dst])  (normalize later)
__global__ void k_edge_scatter(const int* __restrict__ src, const int* __restrict__ dst,
                               const float* __restrict__ Y, const float* __restrict__ e_in,
                               const unsigned* __restrict__ m_enc,
                               float* __restrict__ ssum, float* __restrict__ out) {
    size_t gid  = (size_t)blockIdx.x * blockDim.x + threadIdx.x;
    int    edge = (int)(gid >> 5);
    int    lane = threadIdx.x & 31;
    if (edge >= N_EDGES) return;
    int s = src[edge], d = dst[edge];
    float ex = __expf(e_in[edge] - dec_f32(m_enc[d]));
    if (lane == 0) atomicAdd(&ssum[d], ex);
    float4 v4 = ((const float4*)(Y + (size_t)s * QKV_DIM + 128))[lane]; // v @ 128
    float* op = out + (size_t)d * OUT_DIM + lane * 4;
    atomicAdd(op + 0, ex * v4.x);
    atomicAdd(op + 1, ex * v4.y);
    atomicAdd(op + 2, ex * v4.z);
    atomicAdd(op + 3, ex * v4.w);
}

// ---------------- kernel 7: divide by segment sum ---------------------------
__global__ void k_finalize(float* __restrict__ out, const float* __restrict__ ssum) {
    size_t i = (size_t)blockIdx.x * blockDim.x + threadIdx.x;
    if (i >= (size_t)N_NODES * OUT_DIM) return;
    float s = ssum[i >> 7];                    // OUT_DIM == 128
    out[i] = (s > 0.f) ? out[i] / s : 0.f;
}

extern "C" void kernel_launch(void* const* d_in, const int* in_sizes, int n_in,
                              void* d_out, int out_size, void* d_ws, size_t ws_size,
                              hipStream_t stream) {
    const float* feat  = (const float*)d_in[0];
    const int*   src   = (const int*)  d_in[1];
    const int*   dst   = (const int*)  d_in[2];
    const float* gamma = (const float*)d_in[3];
    const float* beta  = (const float*)d_in[4];
    const float* Wq    = (const float*)d_in[5];
    const float* bq    = (const float*)d_in[6];
    const float* Wk    = (const float*)d_in[7];
    const float* Wv    = (const float*)d_in[8];
    const float* We    = (const float*)d_in[9];
    float*       out   = (float*)d_out;

    // workspace layout (floats): [colsum 128][colsumsq 128][scale 128][shift 128]
    //                            [ssum N][m_enc N][e NE][Y N*256]  ~= 110 MB
    float*    ws       = (float*)d_ws;
    float*    colsum   = ws;
    float*    colsumsq = ws + 128;
    float*    scale    = ws + 256;
    float*    shift    = ws + 384;
    float*    ssum     = ws + 512;
    unsigned* m_enc    = (unsigned*)(ws + 512 + N_NODES);
    float*    e_buf    = ws + 512 + 2 * N_NODES;
    float*    Y        = ws + 512 + 2 * N_NODES + N_EDGES;

    k_init<<<50000, 256, 0, stream>>>(out, ws);
    k_colstats<<<256, 128, 0, stream>>>(feat, colsum, colsumsq);
    k_stats_fin<<<1, 128, 0, stream>>>(colsum, colsumsq, gamma, beta, scale, shift);
    k_qkv_wmma<<<N_NODES / 16, 512, 0, stream>>>(feat, scale, shift, Wq, bq, Wk, Wv, Y);
    k_edge_logits<<<(N_EDGES + 255) / 256, 256, 0, stream>>>(src, dst, Y, We, e_buf, m_enc);
    k_edge_scatter<<<(N_EDGES * 32 + 255) / 256, 256, 0, stream>>>(src, dst, Y, e_buf,
                                                                   m_enc, ssum, out);
    k_finalize<<<(N_NODES * OUT_DIM + 255) / 256, 256, 0, stream>>>(out, ssum);
}